// TopKTopPSampler_82764019794053
// MI455X (gfx1250) — compile-verified
//
#include <hip/hip_runtime.h>
#include <stdint.h>

#define THREADS 256
#define NB 2048

#if defined(__gfx1250__) && __has_builtin(__builtin_amdgcn_global_load_async_to_lds_b128) && __has_builtin(__builtin_amdgcn_s_wait_asynccnt)
#define USE_ASYNC 1
#else
#define USE_ASYNC 0
#endif

// Monotone map: float -> uint32 such that key order == float order (no NaNs in input).
__device__ __forceinline__ unsigned f2key(float x) {
    unsigned u = __float_as_uint(x);
    return (u & 0x80000000u) ? ~u : (u | 0x80000000u);
}

// Stage one 16B-per-thread chunk of a tile: async DMA global->LDS on gfx1250.
__device__ __forceinline__ void stage_tile(const float* gsrc, float* ldst, int tid) {
#if USE_ASYNC
    typedef int v4i __attribute__((vector_size(16)));
    typedef __attribute__((address_space(1))) v4i gv4i_t;
    typedef __attribute__((address_space(3))) v4i lv4i_t;
    __builtin_amdgcn_global_load_async_to_lds_b128(
        (gv4i_t*)(uintptr_t)(gsrc + tid * 4),
        (lv4i_t*)(uintptr_t)(ldst + tid * 4),
        0, 0);
#else
    float4 v = *reinterpret_cast<const float4*>(gsrc + tid * 4);
    *reinterpret_cast<float4*>(ldst + tid * 4) = v;
#endif
}

__device__ __forceinline__ void wait_async_le1() {
#if USE_ASYNC
    __builtin_amdgcn_s_wait_asynccnt(1);
#endif
}
__device__ __forceinline__ void wait_async_le0() {
#if USE_ASYNC
    __builtin_amdgcn_s_wait_asynccnt(0);
#endif
}

// Parallel rank selection: find bucket containing ascending rank r over cnt[0..THREADS*PER).
// All threads participate. Result broadcast via *outB (bucket) / *outR (rank within bucket).
template <int PER>
__device__ __forceinline__ void sel_count(const unsigned* cnt, unsigned r,
                                          unsigned* scan, unsigned* outB, unsigned* outR,
                                          int tid) {
    const int base = tid * PER;
    unsigned part = 0u;
#pragma unroll
    for (int j = 0; j < PER; ++j) part += cnt[base + j];
    scan[tid] = part;
    __syncthreads();
    for (int off = 1; off < THREADS; off <<= 1) {   // Hillis-Steele inclusive scan
        unsigned mine = scan[tid];
        unsigned add = (tid >= off) ? scan[tid - off] : 0u;
        __syncthreads();
        scan[tid] = mine + add;
        __syncthreads();
    }
    const unsigned incl = scan[tid];
    const unsigned excl = incl - part;
    const unsigned total = scan[THREADS - 1];
    const unsigned rr = (r < total) ? r : (total - 1u);   // total >= 1 by construction
    if (rr >= excl && rr < incl) {                        // unique owner
        unsigned cum = excl;
#pragma unroll
        for (int j = 0; j < PER; ++j) {
            unsigned c = cnt[base + j];
            if (cum + c > rr) { *outB = (unsigned)(base + j); *outR = rr - cum; break; }
            cum += c;
        }
    }
    __syncthreads();
}

// Parallel mass selection: find first bucket where cumulative mass exceeds T.
// If frac, T = Tin * total(mass). Result via *outB / *outT (residual target in bucket).
template <int PER>
__device__ __forceinline__ void sel_mass(const float* mass, float Tin, bool frac,
                                         float* scan, int* flag, int* fb,
                                         unsigned* outB, float* outT, int tid) {
    const int base = tid * PER;
    float part = 0.0f;
#pragma unroll
    for (int j = 0; j < PER; ++j) part += mass[base + j];
    scan[tid] = part;
    if (tid == 0) { *flag = 0; *fb = -1; }
    __syncthreads();
    for (int off = 1; off < THREADS; off <<= 1) {
        float mine = scan[tid];
        float add = (tid >= off) ? scan[tid - off] : 0.0f;
        __syncthreads();
        scan[tid] = mine + add;
        __syncthreads();
    }
    const float incl = scan[tid];
    const float excl = incl - part;
    const float total = scan[THREADS - 1];
    const float T = frac ? (Tin * total) : Tin;
    if (T >= excl && T < incl) {                          // unique owner (disjoint intervals)
        int sel = -1; float res = 0.0f; float cum = excl;
#pragma unroll
        for (int j = 0; j < PER; ++j) {
            float mb = mass[base + j];
            if (cum + mb > T) { sel = base + j; res = T - cum; break; }
            cum += mb;
        }
        if (sel < 0) {                                    // rounding safeguard
#pragma unroll
            for (int j = PER - 1; j >= 0; --j)
                if (mass[base + j] != 0.0f) { sel = base + j; res = mass[base + j]; break; }
        }
        if (sel >= 0) { *outB = (unsigned)sel; *outT = res; *flag = 1; }
    }
    __syncthreads();
    if (*flag == 0) {       // uniform branch: T >= total -> pick last nonempty bucket
        int last = -1;
#pragma unroll
        for (int j = PER - 1; j >= 0; --j)
            if (mass[base + j] != 0.0f) { last = base + j; break; }
        if (last >= 0) atomicMax(fb, last);
        __syncthreads();
        if (tid == 0) {
            int b = (*fb >= 0) ? *fb : 0;
            *outB = (unsigned)b; *outT = mass[b];
        }
    }
    __syncthreads();
}

__global__ __launch_bounds__(THREADS)
void topk_topp_sample_kernel(const float* __restrict__ logits,
                             const long long* __restrict__ kArr,
                             const float* __restrict__ pArr,
                             const float* __restrict__ noise,
                             long long* __restrict__ out,
                             int vocab)
{
    __shared__ unsigned s_cnt[NB];
    __shared__ float    s_mass[NB];
    __shared__ float    s_redf[THREADS];
    __shared__ int      s_redi[THREADS];
    __shared__ unsigned s_scanU[THREADS];
    __shared__ float    s_scanF[THREADS];
    __shared__ float    s_tile[2][THREADS * 4];
    __shared__ float    s_m;
    __shared__ unsigned s_selB, s_selR;
    __shared__ float    s_selT;
    __shared__ int      s_flag, s_fb;

    const int row = blockIdx.x;
    const int tid = threadIdx.x;
    const float* grow = logits + (size_t)row * (size_t)vocab;
    const float* nrow = noise  + (size_t)row * (size_t)vocab;
    const int n4 = ((vocab & 3) == 0) ? (vocab >> 2) : 0;   // b128 path only if rows stay aligned
    const float4* g4  = reinterpret_cast<const float4*>(grow);
    const float4* nz4 = reinterpret_cast<const float4*>(nrow);

    // ---- Pass 1: row max (b128 loads) ----
    float mloc = -3.4e38f;
    for (int i = tid; i < n4; i += THREADS) {
        float4 v = g4[i];
        mloc = fmaxf(mloc, fmaxf(fmaxf(v.x, v.y), fmaxf(v.z, v.w)));
    }
    for (int i = (n4 << 2) + tid; i < vocab; i += THREADS) mloc = fmaxf(mloc, grow[i]);
    s_redf[tid] = mloc;
    __syncthreads();
    for (int s = THREADS >> 1; s > 0; s >>= 1) {
        if (tid < s) s_redf[tid] = fmaxf(s_redf[tid], s_redf[tid + s]);
        __syncthreads();
    }
    if (tid == 0) s_m = s_redf[0];
    __syncthreads();
    const float m = s_m;

    // ---- Pass 2: level-0 histogram (count + mass) over top 11 key bits, async-staged tiles ----
    for (int i = tid; i < NB; i += THREADS) { s_cnt[i] = 0u; s_mass[i] = 0.0f; }
    __syncthreads();

    const int TILE = THREADS * 4;
    const int NT = ((vocab & 3) == 0) ? (vocab / TILE) : 0;
    if (NT > 0) stage_tile(grow, s_tile[0], tid);
    for (int t = 0; t < NT; ++t) {
        const int buf = t & 1;
        if (t + 1 < NT) {
            stage_tile(grow + (size_t)(t + 1) * TILE, s_tile[buf ^ 1], tid);
            wait_async_le1();
        } else {
            wait_async_le0();
        }
        __syncthreads();
#pragma unroll
        for (int j = 0; j < 4; ++j) {
            float x = s_tile[buf][tid + j * THREADS];
            unsigned key = f2key(x);
            atomicAdd(&s_cnt[key >> 21], 1u);
            atomicAdd(&s_mass[key >> 21], __expf(x - m));
        }
        __syncthreads();
    }
    for (int i = NT * TILE + tid; i < vocab; i += THREADS) {
        float x = grow[i];
        unsigned key = f2key(x);
        atomicAdd(&s_cnt[key >> 21], 1u);
        atomicAdd(&s_mass[key >> 21], __expf(x - m));
    }
    __syncthreads();

    // ---- Level-0 selection (parallel walks; Z = scan total) ----
    unsigned r = (unsigned)(vocab - (int)kArr[row]);     // ascending rank of top-k cutoff
    sel_count<NB / THREADS>(s_cnt, r, s_scanU, &s_selB, &s_selR, tid);
    unsigned prefK = s_selB; r = s_selR;
    sel_mass<NB / THREADS>(s_mass, 1.0f - pArr[row], true, s_scanF, &s_flag, &s_fb, &s_selB, &s_selT, tid);
    unsigned prefP = s_selB; float T = s_selT;

    // ---- Pass 3: level-1 (key bits 20..10), dual-filtered histograms ----
    for (int i = tid; i < NB; i += THREADS) { s_cnt[i] = 0u; s_mass[i] = 0.0f; }
    __syncthreads();
    for (int i = tid; i < n4; i += THREADS) {
        float4 v = g4[i];
        const float xs[4] = {v.x, v.y, v.z, v.w};
#pragma unroll
        for (int j = 0; j < 4; ++j) {
            unsigned key = f2key(xs[j]);
            if ((key >> 21) == prefK) atomicAdd(&s_cnt[(key >> 10) & 0x7FFu], 1u);
            if ((key >> 21) == prefP) atomicAdd(&s_mass[(key >> 10) & 0x7FFu], __expf(xs[j] - m));
        }
    }
    for (int i = (n4 << 2) + tid; i < vocab; i += THREADS) {
        float x = grow[i];
        unsigned key = f2key(x);
        if ((key >> 21) == prefK) atomicAdd(&s_cnt[(key >> 10) & 0x7FFu], 1u);
        if ((key >> 21) == prefP) atomicAdd(&s_mass[(key >> 10) & 0x7FFu], __expf(x - m));
    }
    __syncthreads();
    sel_count<NB / THREADS>(s_cnt, r, s_scanU, &s_selB, &s_selR, tid);
    prefK = (prefK << 11) | s_selB; r = s_selR;
    sel_mass<NB / THREADS>(s_mass, T, false, s_scanF, &s_flag, &s_fb, &s_selB, &s_selT, tid);
    prefP = (prefP << 11) | s_selB; T = s_selT;

    // ---- Pass 4: level-2 (key bits 9..0, 1024 buckets) -> exact cutoff keys ----
    for (int i = tid; i < NB; i += THREADS) { s_cnt[i] = 0u; s_mass[i] = 0.0f; }
    __syncthreads();
    for (int i = tid; i < n4; i += THREADS) {
        float4 v = g4[i];
        const float xs[4] = {v.x, v.y, v.z, v.w};
#pragma unroll
        for (int j = 0; j < 4; ++j) {
            unsigned key = f2key(xs[j]);
            if ((key >> 10) == prefK) atomicAdd(&s_cnt[key & 0x3FFu], 1u);
            if ((key >> 10) == prefP) atomicAdd(&s_mass[key & 0x3FFu], __expf(xs[j] - m));
        }
    }
    for (int i = (n4 << 2) + tid; i < vocab; i += THREADS) {
        float x = grow[i];
        unsigned key = f2key(x);
        if ((key >> 10) == prefK) atomicAdd(&s_cnt[key & 0x3FFu], 1u);
        if ((key >> 10) == prefP) atomicAdd(&s_mass[key & 0x3FFu], __expf(x - m));
    }
    __syncthreads();
    sel_count<1024 / THREADS>(s_cnt, r, s_scanU, &s_selB, &s_selR, tid);
    const unsigned keyK = (prefK << 10) | s_selB;
    sel_mass<1024 / THREADS>(s_mass, T, false, s_scanF, &s_flag, &s_fb, &s_selB, &s_selT, tid);
    const unsigned keyP = (prefP << 10) | s_selB;
    const unsigned cut = (keyK > keyP) ? keyK : keyP;

    // ---- Pass 5: weighted argmax over kept tokens: argmax exp(l-m)/noise, first-index ties ----
    float best = -1.0f; int bidx = 0;
    for (int i = tid; i < n4; i += THREADS) {
        float4 x = g4[i];
        float4 nv = nz4[i];
        const float xs[4] = {x.x, x.y, x.z, x.w};
        const float ns[4] = {nv.x, nv.y, nv.z, nv.w};
#pragma unroll
        for (int j = 0; j < 4; ++j) {
            if (f2key(xs[j]) >= cut) {
                float sc = __expf(xs[j] - m) / ns[j];
                if (sc > best) { best = sc; bidx = (i << 2) + j; }
            }
        }
    }
    for (int i = (n4 << 2) + tid; i < vocab; i += THREADS) {
        float x = grow[i];
        if (f2key(x) >= cut) {
            float sc = __expf(x - m) / nrow[i];
            if (sc > best) { best = sc; bidx = i; }
        }
    }
    s_redf[tid] = best; s_redi[tid] = bidx;
    __syncthreads();
    for (int s = THREADS >> 1; s > 0; s >>= 1) {
        if (tid < s) {
            float ov = s_redf[tid + s]; int oi = s_redi[tid + s];
            float mv = s_redf[tid];     int mi = s_redi[tid];
            if (ov > mv || (ov == mv && oi < mi)) { s_redf[tid] = ov; s_redi[tid] = oi; }
        }
        __syncthreads();
    }
    if (tid == 0) out[row] = (long long)s_redi[0];
}

extern "C" void kernel_launch(void* const* d_in, const int* in_sizes, int n_in,
                              void* d_out, int out_size, void* d_ws, size_t ws_size,
                              hipStream_t stream) {
    (void)n_in; (void)d_ws; (void)ws_size; (void)out_size;
    const float*     logits = (const float*)d_in[0];
    const long long* k      = (const long long*)d_in[1];
    const float*     p      = (const float*)d_in[2];
    const float*     noise  = (const float*)d_in[3];
    const int rows  = in_sizes[1];              // BATCH (size of k)
    const int vocab = in_sizes[0] / rows;       // VOCAB
    topk_topp_sample_kernel<<<rows, THREADS, 0, stream>>>(
        logits, k, p, noise, (long long*)d_out, vocab);
}